// GAT_LSTM_Model_5514738008802
// MI455X (gfx1250) — compile-verified
//
#include <hip/hip_runtime.h>

typedef __attribute__((ext_vector_type(2))) float v2f;
typedef __attribute__((ext_vector_type(8))) float v8f;

static constexpr int NN = 50000;     // nodes
static constexpr int EE = 1600000;   // edges (without self loops)
static constexpr int ET = EE + NN;   // edges + self loops

// ---------- helpers ----------
__device__ __forceinline__ unsigned fenc(float f) {
  unsigned u = __float_as_uint(f);
  return (u & 0x80000000u) ? ~u : (u | 0x80000000u);
}
__device__ __forceinline__ float fdec(unsigned u) {
  return (u & 0x80000000u) ? __uint_as_float(u & 0x7FFFFFFFu)
                           : __uint_as_float(~u);
}
__device__ __forceinline__ float sigmoidf_(float x) {
  return 1.0f / (1.0f + __expf(-x));
}

// ---------- WMMA f32 GEMM: one wave computes one 16x16 tile of C ----------
// C[M x Ncol] = A[M x K] * B   (BT=0: B is K x Ncol row-major; BT=1: B is Ncol x K row-major)
// ACC=1: accumulate into existing C.
template<bool BT, bool ACC>
__global__ __launch_bounds__(32)
void gemm16(const float* __restrict__ A, int lda,
            const float* __restrict__ B, int ldb,
            float* __restrict__ C, int ldc, int K)
{
  const int lane = threadIdx.x & 31;
  const int m    = lane & 15;           // row within tile (A) / col within tile (B,C)
  const int kb   = (lane >> 4) << 1;    // K sub-offset: lanes 0-15 -> K0/K1, 16-31 -> K2/K3
  const int row0 = blockIdx.x << 4;
  const int col0 = blockIdx.y << 4;
  const int rbase = row0 + ((lane >> 4) << 3);

  v8f c = {};
  if (ACC) {
#pragma unroll
    for (int v = 0; v < 8; ++v)
      c[v] = C[(size_t)(rbase + v) * ldc + col0 + m];
  }

  for (int kk = 0; kk < K; kk += 4) {
    v2f a, b;
    a.x = A[(size_t)(row0 + m) * lda + kk + kb];
    a.y = A[(size_t)(row0 + m) * lda + kk + kb + 1];
    if (BT) {
      b.x = B[(size_t)(col0 + m) * ldb + kk + kb];
      b.y = B[(size_t)(col0 + m) * ldb + kk + kb + 1];
    } else {
      b.x = B[(size_t)(kk + kb) * ldb + col0 + m];
      b.y = B[(size_t)(kk + kb + 1) * ldb + col0 + m];
    }
    c = __builtin_amdgcn_wmma_f32_16x16x4_f32(false, a, false, b,
                                              (short)0, c, false, false);
  }

#pragma unroll
  for (int v = 0; v < 8; ++v)
    C[(size_t)(rbase + v) * ldc + col0 + m] = c[v];
}

// ---------- GAT elementwise kernels ----------
__global__ void init_layer(float* __restrict__ acc, unsigned* __restrict__ emaxu,
                           float* __restrict__ den, int accN, int nh)
{
  int t = blockIdx.x * blockDim.x + threadIdx.x;
  if (t < accN) acc[t] = 0.0f;
  if (t < nh) { den[t] = 0.0f; emaxu[t] = 0x007FFFFFu; }  // enc(-inf)
}

__global__ void al_kernel(const float* __restrict__ xw,
                          const float* __restrict__ asrc,
                          const float* __restrict__ adst,
                          float* __restrict__ als, float* __restrict__ ald,
                          int H, int C)
{
  int t = blockIdx.x * blockDim.x + threadIdx.x;
  if (t >= NN * H) return;
  int n = t / H, h = t - n * H;
  const float* row = xw + (size_t)n * H * C + (size_t)h * C;
  const float* as  = asrc + h * C;
  const float* ad  = adst + h * C;
  float s = 0.f, d = 0.f;
  for (int c = 0; c < C; ++c) { float v = row[c]; s += v * as[c]; d += v * ad[c]; }
  als[t] = s; ald[t] = d;
}

__global__ void edge_logits(const int* __restrict__ ei,
                            const float* __restrict__ als,
                            const float* __restrict__ ald,
                            float* __restrict__ ebuf,
                            unsigned* __restrict__ emaxu,
                            int H, int hshift)
{
  int t = blockIdx.x * blockDim.x + threadIdx.x;
  if (t >= (ET << hshift)) return;
  int e = t >> hshift, h = t & (H - 1);
  int s = (e < EE) ? ei[e]      : (e - EE);
  int d = (e < EE) ? ei[EE + e] : (e - EE);
  float v = als[s * H + h] + ald[d * H + h];
  v = (v > 0.f) ? v : 0.2f * v;              // leaky_relu(0.2)
  ebuf[t] = v;
  atomicMax(emaxu + d * H + h, fenc(v));
}

__global__ void emax_fin(unsigned* __restrict__ emaxu, int n)
{
  int t = blockIdx.x * blockDim.x + threadIdx.x;
  if (t >= n) return;
  float f = fdec(emaxu[t]);
  if (!(fabsf(f) < 3.0e38f)) f = 0.0f;       // isfinite -> else 0
  ((float*)emaxu)[t] = f;
}

__global__ void edge_exp(const int* __restrict__ ei,
                         const float* __restrict__ emaxf,
                         float* __restrict__ ebuf,
                         float* __restrict__ den,
                         int H, int hshift)
{
  int t = blockIdx.x * blockDim.x + threadIdx.x;
  if (t >= (ET << hshift)) return;
  int e = t >> hshift, h = t & (H - 1);
  int d = (e < EE) ? ei[EE + e] : (e - EE);
  float ex = __expf(ebuf[t] - emaxf[d * H + h]);
  ebuf[t] = ex;
  atomicAdd(den + d * H + h, ex);
}

__global__ void edge_agg(const int* __restrict__ ei,
                         const float* __restrict__ ebuf,
                         const float* __restrict__ den,
                         const float* __restrict__ xw,
                         float* __restrict__ acc,
                         int H, int hcshift, int cshift)
{
  int t = blockIdx.x * blockDim.x + threadIdx.x;
  if (t >= (ET << hcshift)) return;
  int e   = t >> hcshift;
  int idx = t & ((1 << hcshift) - 1);
  int h   = idx >> cshift;
  int s = (e < EE) ? ei[e]      : (e - EE);
  int d = (e < EE) ? ei[EE + e] : (e - EE);
  float alpha = ebuf[(size_t)e * H + h] / (den[d * H + h] + 1e-16f);
  atomicAdd(acc + (((size_t)d) << hcshift) + idx,
            alpha * xw[(((size_t)s) << hcshift) + idx]);
}

__global__ void bias_act(const float* __restrict__ acc, const float* __restrict__ b,
                         float* __restrict__ out, int hc, int elu)
{
  int t = blockIdx.x * blockDim.x + threadIdx.x;
  if (t >= NN * hc) return;
  float v = acc[t] + b[t % hc];
  if (elu) v = (v > 0.f) ? v : (__expf(v) - 1.0f);
  out[t] = v;
}

// ---------- LSTM kernels ----------
__global__ void gather_rows(const float* __restrict__ src, const int* __restrict__ nid,
                            float* __restrict__ dst)
{
  int t = blockIdx.x * blockDim.x + threadIdx.x;
  if (t >= NN * 64) return;
  int r = t >> 6, c = t & 63;
  dst[t] = src[((size_t)nid[r] << 6) + c];
}

__global__ void lstm_cell(const float* __restrict__ gates,
                          const float* __restrict__ bih, const float* __restrict__ bhh,
                          const float* __restrict__ cfull_l, const int* __restrict__ nid,
                          float* __restrict__ hs, float* __restrict__ cs)
{
  int t = blockIdx.x * blockDim.x + threadIdx.x;
  if (t >= NN * 64) return;
  int n = t >> 6, j = t & 63;
  const float* gr = gates + (size_t)n * 256;
  float xi = gr[j]        + bih[j]        + bhh[j];
  float xf = gr[64 + j]   + bih[64 + j]   + bhh[64 + j];
  float xg = gr[128 + j]  + bih[128 + j]  + bhh[128 + j];
  float xo = gr[192 + j]  + bih[192 + j]  + bhh[192 + j];
  float i = sigmoidf_(xi), f = sigmoidf_(xf), o = sigmoidf_(xo);
  float g = tanhf(xg);
  float c0 = cfull_l[((size_t)nid[n] << 6) + j];
  float cn = f * c0 + i * g;
  float hn = o * tanhf(cn);
  cs[t] = cn;
  hs[t] = hn;
}

__global__ void pred_kernel(const float* __restrict__ h, const float* __restrict__ W,
                            const float* __restrict__ b, float* __restrict__ pred)
{
  int t = blockIdx.x * blockDim.x + threadIdx.x;
  if (t >= NN) return;
  float s = b[0];
#pragma unroll
  for (int j = 0; j < 64; ++j) s += h[((size_t)t << 6) + j] * W[j];
  pred[t] = s;
}

// ---------- host orchestration ----------
static void gat_layer(const float* feat, int K, const float* W,
                      const float* asrc, const float* adst, const float* bias,
                      int H, int hshift, int C, int cshift, bool elu,
                      const int* ei, float* xw, float* acc, float* out,
                      float* als, float* ald, float* emaxb, float* den, float* ebuf,
                      hipStream_t st)
{
  const int HC = H * C, hcshift = hshift + cshift;
  dim3 gg(NN / 16, HC / 16);
  gemm16<false, false><<<gg, 32, 0, st>>>(feat, K, W, HC, xw, HC, K);
  const int nh = NN * H;
  al_kernel<<<(nh + 255) / 256, 256, 0, st>>>(xw, asrc, adst, als, ald, H, C);
  const int accN = NN * HC;
  init_layer<<<(accN + 255) / 256, 256, 0, st>>>(acc, (unsigned*)emaxb, den, accN, nh);
  const int eth = ET << hshift;
  edge_logits<<<(eth + 255) / 256, 256, 0, st>>>(ei, als, ald, ebuf, (unsigned*)emaxb, H, hshift);
  emax_fin<<<(nh + 255) / 256, 256, 0, st>>>((unsigned*)emaxb, nh);
  edge_exp<<<(eth + 255) / 256, 256, 0, st>>>(ei, emaxb, ebuf, den, H, hshift);
  const int etc_ = ET << hcshift;
  edge_agg<<<(etc_ + 255) / 256, 256, 0, st>>>(ei, ebuf, den, xw, acc, H, hcshift, cshift);
  bias_act<<<(accN + 255) / 256, 256, 0, st>>>(acc, bias, out, HC, elu ? 1 : 0);
}

static void lstm_layer(const float* inp, const float* hprev,
                       const float* Wih, const float* Whh,
                       const float* bih, const float* bhh,
                       const float* cfull_l, const int* nid,
                       float* gates, float* hs, float* cs, hipStream_t st)
{
  dim3 gg(NN / 16, 256 / 16);
  gemm16<true, false><<<gg, 32, 0, st>>>(inp,   64, Wih, 64, gates, 256, 64);
  gemm16<true, true ><<<gg, 32, 0, st>>>(hprev, 64, Whh, 64, gates, 256, 64);
  const int n = NN * 64;
  lstm_cell<<<(n + 255) / 256, 256, 0, st>>>(gates, bih, bhh, cfull_l, nid, hs, cs);
}

extern "C" void kernel_launch(void* const* d_in, const int* in_sizes, int n_in,
                              void* d_out, int out_size, void* d_ws, size_t ws_size,
                              hipStream_t stream)
{
  (void)in_sizes; (void)n_in; (void)out_size; (void)ws_size;

  const float* x      = (const float*)d_in[0];
  const int*   ei     = (const int*)  d_in[1];   // [2,E]: row0=src, row1=dst
  // d_in[2] = edge_attr (unused by reference)
  const int*   nid    = (const int*)  d_in[3];
  const float* h_full = (const float*)d_in[4];
  const float* c_full = (const float*)d_in[5];
  const float* W0 = (const float*)d_in[6],  *as0 = (const float*)d_in[7];
  const float* ad0 = (const float*)d_in[8], *b0 = (const float*)d_in[9];
  const float* W1 = (const float*)d_in[10], *as1 = (const float*)d_in[11];
  const float* ad1 = (const float*)d_in[12], *b1 = (const float*)d_in[13];
  const float* W2 = (const float*)d_in[14], *as2 = (const float*)d_in[15];
  const float* ad2 = (const float*)d_in[16], *b2 = (const float*)d_in[17];
  const float* Wih0 = (const float*)d_in[18], *Whh0 = (const float*)d_in[19];
  const float* bih0 = (const float*)d_in[20], *bhh0 = (const float*)d_in[21];
  const float* Wih1 = (const float*)d_in[22], *Whh1 = (const float*)d_in[23];
  const float* bih1 = (const float*)d_in[24], *bhh1 = (const float*)d_in[25];
  const float* outW = (const float*)d_in[26], *outb = (const float*)d_in[27];

  // workspace layout (floats)
  float* ws    = (float*)d_ws;
  float* xw    = ws;                           // NN*128
  float* acc   = xw  + (size_t)NN * 128;       // NN*128
  float* hbuf  = acc + (size_t)NN * 128;       // NN*128  (h1 -> h2 -> hprev)
  float* gbuf  = hbuf + (size_t)NN * 128;      // NN*64
  float* als   = gbuf + (size_t)NN * 64;       // NN*4
  float* ald   = als + (size_t)NN * 4;         // NN*4
  float* emaxb = ald + (size_t)NN * 4;         // NN*4 (uint then float)
  float* den   = emaxb + (size_t)NN * 4;       // NN*4
  float* ebuf  = den + (size_t)NN * 4;         // ET*4
  float* gates = xw;                           // reuse xw+acc region: NN*256
  float* hprev = hbuf;                         // reuse after GAT

  // output layout: [pred NN][hs 2*NN*64][cs 2*NN*64]
  float* outp = (float*)d_out;
  float* pred = outp;
  float* hs0  = outp + NN;
  float* hs1  = hs0 + (size_t)NN * 64;
  float* cs0  = hs1 + (size_t)NN * 64;
  float* cs1  = cs0 + (size_t)NN * 64;

  // GAT stack
  gat_layer(x,    32,  W0, as0, ad0, b0, 4, 2, 32, 5, true,  ei, xw, acc, hbuf,
            als, ald, emaxb, den, ebuf, stream);                       // h1
  gat_layer(hbuf, 128, W1, as1, ad1, b1, 4, 2, 32, 5, true,  ei, xw, acc, hbuf,
            als, ald, emaxb, den, ebuf, stream);                       // h2
  gat_layer(hbuf, 128, W2, as2, ad2, b2, 1, 0, 64, 6, false, ei, xw, acc, gbuf,
            als, ald, emaxb, den, ebuf, stream);                       // g

  // LSTM stack
  const int n64 = NN * 64;
  gather_rows<<<(n64 + 255) / 256, 256, 0, stream>>>(h_full, nid, hprev);
  lstm_layer(gbuf, hprev, Wih0, Whh0, bih0, bhh0, c_full, nid, gates, hs0, cs0, stream);
  gather_rows<<<(n64 + 255) / 256, 256, 0, stream>>>(h_full + (size_t)NN * 64, nid, hprev);
  lstm_layer(hs0, hprev, Wih1, Whh1, bih1, bhh1, c_full + (size_t)NN * 64, nid,
             gates, hs1, cs1, stream);

  pred_kernel<<<(NN + 255) / 256, 256, 0, stream>>>(hs1, outW, outb, pred);
}